// sentiment_visual_60206851555513
// MI455X (gfx1250) — compile-verified
//
#include <hip/hip_runtime.h>
#include <math.h>

// Problem dims
#define Bc 16
#define Sc 2048
#define Rc 196
#define Hc 768
#define ROWTILES 13   // ceil(196/16) row tiles per batch

typedef __attribute__((ext_vector_type(2))) float v2f;
typedef __attribute__((ext_vector_type(8))) float v8f;

// ---------------------------------------------------------------------------
// Kernel A: scores[b,r] = sum_o tanh( img[b,r,:] . W_img[o,:] ) * w2[o]
// 8 waves/block; wave w handles col-tiles [6w, 6w+6); each tile is a
// 16x16 WMMA f32 accumulation over K=768 in steps of 4 (V_WMMA_F32_16X16X4_F32).
// ---------------------------------------------------------------------------
__global__ __launch_bounds__(256) void img_scores_kernel(
    const float* __restrict__ img,    // [B,R,H]
    const float* __restrict__ Wimg,   // [H,H], o-major
    const float* __restrict__ watt,   // [2H]; w2 = watt + H
    float* __restrict__ scores)       // [B,R]
{
    const int b    = blockIdx.x;            // 0..15
    const int rt   = blockIdx.y;            // 0..12
    const int row0 = rt * 16;
    const int tid  = threadIdx.x;
    const int wave = tid >> 5;              // 0..7
    const int lane = tid & 31;
    const int kh   = lane >> 4;             // lane half: selects K pair {0,1}/{2,3}
    const int l16  = lane & 15;

    __shared__ float s_scores[16];
    if (tid < 16) s_scores[tid] = 0.0f;
    __syncthreads();

    // A operand: lane l (both halves) holds row M=l16 -> r = row0+l16 (clamped)
    int r  = row0 + l16;
    int rc = (r < Rc) ? r : (Rc - 1);
    const float* arow = img + ((size_t)b * Rc + rc) * Hc + kh * 2;
    const float* w2   = watt + Hc;

    float partial[8];
#pragma unroll
    for (int j = 0; j < 8; ++j) partial[j] = 0.0f;

    for (int t = 0; t < 6; ++t) {
        const int ct = wave * 6 + t;          // col tile 0..47
        const int o  = ct * 16 + l16;         // B operand: lane holds col N=l16
        const float* brow = Wimg + (size_t)o * Hc + kh * 2;

        v8f acc = {};
#pragma unroll 4
        for (int k = 0; k < Hc; k += 4) {
            v2f a, bb;
            a.x  = arow[k];  a.y  = arow[k + 1];
            bb.x = brow[k];  bb.y = brow[k + 1];
            // D = A(16x4,f32) x B(4x16,f32) + C(16x16,f32)
            acc = __builtin_amdgcn_wmma_f32_16x16x4_f32(
                      false, a, false, bb, (short)0, acc, false, false);
        }
        const float w2v = w2[o];
#pragma unroll
        for (int j = 0; j < 8; ++j)
            partial[j] += tanhf(acc[j]) * w2v;   // fold tanh + w2 dot per lane
    }

    // Reduce over N (16 lanes of each half). xor masks 1..8 stay within halves.
#pragma unroll
    for (int j = 0; j < 8; ++j) {
        float v = partial[j];
        v += __shfl_xor(v, 1, 32);
        v += __shfl_xor(v, 2, 32);
        v += __shfl_xor(v, 4, 32);
        v += __shfl_xor(v, 8, 32);
        partial[j] = v;   // lane 0 -> row j, lane 16 -> row j+8
    }
    if (l16 == 0) {
#pragma unroll
        for (int j = 0; j < 8; ++j)
            atomicAdd(&s_scores[j + kh * 8], partial[j]);   // ds_add_f32
    }
    __syncthreads();

    if (tid < 16) {
        int rr = row0 + tid;
        if (rr < Rc) scores[b * Rc + rr] = s_scores[tid];
    }
}

// ---------------------------------------------------------------------------
// Kernel B: per batch b -> att = softmax_r(scores[b,:]);
//           ctx[b,h] = sum_r att[r] * img[b,r,h]
// (text term + b_att cancel inside softmax: shift invariance.)
// ---------------------------------------------------------------------------
__global__ __launch_bounds__(256) void softmax_ctx_kernel(
    const float* __restrict__ img,     // [B,R,H]
    const float* __restrict__ scores,  // [B,R]
    float* __restrict__ ctx)           // [B,H]
{
    const int b   = blockIdx.x;
    const int tid = threadIdx.x;

    __shared__ float buf[256];
    __shared__ float s_att[Rc];

    float v = (tid < Rc) ? scores[b * Rc + tid] : -INFINITY;
    buf[tid] = v;
    __syncthreads();
    for (int st = 128; st > 0; st >>= 1) {
        if (tid < st) buf[tid] = fmaxf(buf[tid], buf[tid + st]);
        __syncthreads();
    }
    const float mx = buf[0];
    __syncthreads();

    const float e = (tid < Rc) ? expf(v - mx) : 0.0f;
    buf[tid] = e;
    __syncthreads();
    for (int st = 128; st > 0; st >>= 1) {
        if (tid < st) buf[tid] += buf[tid + st];
        __syncthreads();
    }
    const float inv = 1.0f / buf[0];
    if (tid < Rc) s_att[tid] = e * inv;
    __syncthreads();

    for (int h = tid; h < Hc; h += 256) {
        float acc = 0.0f;
        const float* ib = img + ((size_t)b * Rc) * Hc + h;
        for (int rr = 0; rr < Rc; ++rr)
            acc = fmaf(s_att[rr], ib[(size_t)rr * Hc], acc);
        ctx[b * Hc + h] = acc;
    }
}

// ---------------------------------------------------------------------------
// Kernel C: out[b,s,:] = ctx[b,:] for all s (pure bandwidth, float4 stores).
// 192 threads = one H-row of float4; each thread writes 8 consecutive s-rows.
// ---------------------------------------------------------------------------
__global__ __launch_bounds__(192) void broadcast_kernel(
    const float* __restrict__ ctx,   // [B,H]
    float4* __restrict__ out)        // [B*S*H/4]
{
    const int H4  = Hc / 4;                 // 192
    const int b   = blockIdx.y;
    const int s0  = blockIdx.x * 8;         // 8 s-rows per block
    const int tid = threadIdx.x;            // 0..191

    const float4 val = ((const float4*)ctx)[b * H4 + tid];
#pragma unroll
    for (int i = 0; i < 8; ++i)
        out[((size_t)(b * Sc + s0 + i)) * H4 + tid] = val;
}

// ---------------------------------------------------------------------------
extern "C" void kernel_launch(void* const* d_in, const int* in_sizes, int n_in,
                              void* d_out, int out_size, void* d_ws, size_t ws_size,
                              hipStream_t stream) {
    // setup_inputs order: text_features, img_features, W_text, b_text, W_img, w_att, b_att
    // text_features / W_text / b_text / b_att are mathematically dead (softmax shift invariance).
    const float* img  = (const float*)d_in[1];
    const float* Wimg = (const float*)d_in[4];
    const float* watt = (const float*)d_in[5];
    float* out = (float*)d_out;

    float* scores = (float*)d_ws;                          // 16*196*4   = 12.5 KB
    float* ctx    = (float*)((char*)d_ws + 16384);         // 16*768*4   = 48 KB

    img_scores_kernel<<<dim3(Bc, ROWTILES), 256, 0, stream>>>(img, Wimg, watt, scores);
    softmax_ctx_kernel<<<Bc, 256, 0, stream>>>(img, scores, ctx);
    broadcast_kernel<<<dim3(Sc / 8, Bc), 192, 0, stream>>>(ctx, (float4*)out);
}